// LSSViewTransformer_69587060130311
// MI455X (gfx1250) — compile-verified
//
#include <hip/hip_runtime.h>
#include <stdint.h>

// bev_pool_v2 for MI455X (gfx1250): gather depth & feat via ranks, multiply,
// atomic segment-sum into BEV grid. Memory-bound; L2-resident working set
// (feat 43MB + depth 48MB + out 42MB = 133MB < 192MB L2).

#define AS1 __attribute__((address_space(1)))
#define AS3 __attribute__((address_space(3)))

#define BEV_C 80
#define ZYX   16384            // 1 * 128 * 128
#define GROUPS_PER_PT 20       // 80 channels / float4
#define BLK 256

__device__ __forceinline__ void atom_add_f32(float* p, float v) {
#if __has_builtin(__builtin_amdgcn_global_atomic_fadd_f32)
    __builtin_amdgcn_global_atomic_fadd_f32((AS1 float*)p, v);
#else
    atomicAdd(p, v);
#endif
}

__global__ void __launch_bounds__(BLK) bev_pool_kernel(
    const float* __restrict__ depth,
    const float* __restrict__ feat,
    const int*   __restrict__ ranks_depth,
    const int*   __restrict__ ranks_feat,
    const int*   __restrict__ ranks_bev,
    float*       __restrict__ out,
    int NP)
{
    __shared__ int      s_rd[BLK];
    __shared__ int      s_rf[BLK];
    __shared__ int      s_rb[BLK];
    __shared__ float    s_dv[BLK];
    __shared__ uint32_t s_ob[BLK];

    const int tid = threadIdx.x;
    const int p0  = blockIdx.x * BLK;
    const int p   = p0 + tid;
    const int pc  = (p < NP) ? p : (NP - 1);   // clamp: always in-bounds

    // ---- Phase 1: stage ranks into LDS via gfx1250 async load-to-LDS ----
#if __has_builtin(__builtin_amdgcn_global_load_async_to_lds_b32)
    __builtin_amdgcn_global_load_async_to_lds_b32(
        (AS1 int*)(ranks_depth + pc), (AS3 int*)&s_rd[tid], 0, 0);
    __builtin_amdgcn_global_load_async_to_lds_b32(
        (AS1 int*)(ranks_feat + pc),  (AS3 int*)&s_rf[tid], 0, 0);
    __builtin_amdgcn_global_load_async_to_lds_b32(
        (AS1 int*)(ranks_bev + pc),   (AS3 int*)&s_rb[tid], 0, 0);
#if __has_builtin(__builtin_amdgcn_s_wait_asynccnt)
    __builtin_amdgcn_s_wait_asynccnt(0);
#else
    asm volatile("s_wait_asynccnt 0" ::: "memory");
#endif
#else
    s_rd[tid] = ranks_depth[pc];
    s_rf[tid] = ranks_feat[pc];
    s_rb[tid] = ranks_bev[pc];
#endif
    __syncthreads();

    // One depth gather per point (shared by its 20 channel-group threads),
    // and precompute the output base: ((b*80) << 14) | spatial.
    {
        s_dv[tid] = depth[s_rd[tid]];
        int rb = s_rb[tid];
        s_ob[tid] = (((uint32_t)(rb >> 14) * BEV_C) << 14) | (uint32_t)(rb & (ZYX - 1));
    }
    __syncthreads();

    const int nvalid = (NP - p0 < BLK) ? (NP - p0) : BLK;

    // ---- Phase 2: 256 threads sweep 256 points x 20 float4 groups ----
    if (nvalid == BLK) {
        // Fast path: every tile except the last — no per-iteration guard.
#pragma unroll
        for (int it = 0; it < GROUPS_PER_PT; ++it) {
            int w  = it * BLK + tid;
            int pl = w / GROUPS_PER_PT;        // 0..255 point-in-block
            int j  = w - pl * GROUPS_PER_PT;   // 0..19 channel group
            float d = s_dv[pl];
            const float4 f =
                *(const float4*)(feat + (size_t)s_rf[pl] * BEV_C + (j << 2));
            float* o = out + s_ob[pl] + ((uint32_t)(j << 2) << 14);
            atom_add_f32(o,            d * f.x);
            atom_add_f32(o + ZYX,      d * f.y);
            atom_add_f32(o + 2 * ZYX,  d * f.z);
            atom_add_f32(o + 3 * ZYX,  d * f.w);
        }
    } else {
#pragma unroll
        for (int it = 0; it < GROUPS_PER_PT; ++it) {
            int w  = it * BLK + tid;
            int pl = w / GROUPS_PER_PT;
            int j  = w - pl * GROUPS_PER_PT;
            if (pl < nvalid) {
                float d = s_dv[pl];
                const float4 f =
                    *(const float4*)(feat + (size_t)s_rf[pl] * BEV_C + (j << 2));
                float* o = out + s_ob[pl] + ((uint32_t)(j << 2) << 14);
                atom_add_f32(o,            d * f.x);
                atom_add_f32(o + ZYX,      d * f.y);
                atom_add_f32(o + 2 * ZYX,  d * f.z);
                atom_add_f32(o + 3 * ZYX,  d * f.w);
            }
        }
    }
}

__global__ void __launch_bounds__(BLK) bev_zero_kernel(float* __restrict__ out, int n) {
    int i4 = (blockIdx.x * BLK + threadIdx.x) * 4;
    if (i4 + 3 < n) {
        *(float4*)(out + i4) = make_float4(0.f, 0.f, 0.f, 0.f);
    } else {
        for (int k = i4; k < n; ++k) out[k] = 0.f;
    }
}

extern "C" void kernel_launch(void* const* d_in, const int* in_sizes, int n_in,
                              void* d_out, int out_size, void* d_ws, size_t ws_size,
                              hipStream_t stream) {
    const float* depth = (const float*)d_in[0];
    const float* feat  = (const float*)d_in[1];
    const int*   rd    = (const int*)d_in[2];
    const int*   rf    = (const int*)d_in[3];
    const int*   rb    = (const int*)d_in[4];
    float*       out   = (float*)d_out;
    const int NP = in_sizes[2];

    const int zblocks = ((out_size + 3) / 4 + BLK - 1) / BLK;
    bev_zero_kernel<<<zblocks, BLK, 0, stream>>>(out, out_size);

    const int blocks = (NP + BLK - 1) / BLK;
    bev_pool_kernel<<<blocks, BLK, 0, stream>>>(depth, feat, rd, rf, rb, out, NP);
}